// FastGTNs_60309930770873
// MI455X (gfx1250) — compile-verified
//
#include <hip/hip_runtime.h>

#define N_NODES 50000
#define E_EDGES 300000
#define T_TYPES 3
#define C_CH    2
#define HID     64
#define NL      2
#define BETA    0.5f
#define TPROB   0.8f

typedef float v2f __attribute__((ext_vector_type(2)));
typedef float v8f __attribute__((ext_vector_type(8)));

// ---------------------------------------------------------------------------
// Tiny kernel: softmax over T=3 for all (block, layer, channel) combos.
// filt layout: [block(2)][layer(NL)][C][T]
// ---------------------------------------------------------------------------
__global__ void filt_kernel(const float* __restrict__ lw0,
                            const float* __restrict__ lw1,
                            float* __restrict__ filt) {
    int i = threadIdx.x;                 // 0..7 -> b*4 + l*2 + c
    if (i >= 2 * NL * C_CH) return;
    int b = i >> 2, l = (i >> 1) & 1, c = i & 1;
    const float* lw = (b == 0 ? lw0 : lw1) + (l * C_CH + c) * T_TYPES;
    float m  = fmaxf(lw[0], fmaxf(lw[1], lw[2]));
    float e0 = __expf(lw[0] - m), e1 = __expf(lw[1] - m), e2 = __expf(lw[2] - m);
    float s  = e0 + e1 + e2;
    float* o = filt + i * T_TYPES;
    o[0] = e0 / s; o[1] = e1 / s; o[2] = e2 / s;
}

// ---------------------------------------------------------------------------
// fp32 WMMA GEMM: Out[c] = A @ B[c]  (Out is M x 64, B is K x 64, A is M x K)
// grid.y = channel; optional second copy (Out2), optional bias + relu.
// One wave32 computes a 16x64 tile using V_WMMA_F32_16X16X4_F32.
// ---------------------------------------------------------------------------
__global__ __launch_bounds__(256) void gemm_n64_f32_wmma(
    const float* __restrict__ A, int lda,
    const float* __restrict__ Bmat, long strideB,
    float* __restrict__ Out, float* __restrict__ Out2, long strideO,
    const float* __restrict__ bias, int relu, int M, int K)
{
    const int c = blockIdx.y;
    const float* B = Bmat + (long)c * strideB;
    float* O  = Out + (long)c * strideO;
    float* O2 = Out2 ? (Out2 + (long)c * strideO) : nullptr;

    const int wave = (int)((blockIdx.x * blockDim.x + threadIdx.x) >> 5);
    const int row0 = wave * 16;
    if (row0 >= M) return;                 // wave-uniform: EXEC stays all-1s
    const int lane = threadIdx.x & 31;
    const int half = lane >> 4;            // which 16-lane half
    const int m    = lane & 15;            // row (A) / col (B,D) within tile

    v8f acc0 = {}; v8f acc1 = {}; v8f acc2 = {}; v8f acc3 = {};

    const float* Arow = A + (long)(row0 + m) * lda + 2 * half;
    #pragma unroll 4
    for (int k0 = 0; k0 < K; k0 += 4) {
        // A fragment 16x4: VGPR0 = K(k0+2*half), VGPR1 = K(k0+2*half+1)
        v2f a = *(const v2f*)(Arow + k0);
        const int kb = k0 + 2 * half;
        const float* Bp0 = B + (long)kb * 64 + m;   // row kb
        const float* Bp1 = Bp0 + 64;                // row kb+1
        v2f b0 = { Bp0[ 0], Bp1[ 0] };
        v2f b1 = { Bp0[16], Bp1[16] };
        v2f b2 = { Bp0[32], Bp1[32] };
        v2f b3 = { Bp0[48], Bp1[48] };
        acc0 = __builtin_amdgcn_wmma_f32_16x16x4_f32(false, a, false, b0, (short)0, acc0, false, false);
        acc1 = __builtin_amdgcn_wmma_f32_16x16x4_f32(false, a, false, b1, (short)0, acc1, false, false);
        acc2 = __builtin_amdgcn_wmma_f32_16x16x4_f32(false, a, false, b2, (short)0, acc2, false, false);
        acc3 = __builtin_amdgcn_wmma_f32_16x16x4_f32(false, a, false, b3, (short)0, acc3, false, false);
    }

    v8f accs[4] = { acc0, acc1, acc2, acc3 };
    #pragma unroll
    for (int nt = 0; nt < 4; ++nt) {
        const int col = nt * 16 + m;
        const float bv = bias ? bias[col] : 0.0f;
        #pragma unroll
        for (int r = 0; r < 8; ++r) {
            const int row = row0 + r + 8 * half;    // D layout: VGPR r -> M=r / r+8
            float v = accs[nt][r] + bv;
            if (relu) v = fmaxf(v, 0.0f);
            O[(long)row * 64 + col] = v;
            if (O2) O2[(long)row * 64 + col] = v;
        }
    }
}

// ---------------------------------------------------------------------------
// SpMM-mix: Hout[c][row] += val * filt[c][etype] * H[c][col]   (Hout pre-zeroed)
// 16 lanes per (edge, channel); float4 gather; hw fp32 atomics (L2-resident).
// ---------------------------------------------------------------------------
__global__ __launch_bounds__(256) void spmm_kernel(
    const float* __restrict__ H, float* __restrict__ Hout,
    const int* __restrict__ ei0, const float* __restrict__ ev0,
    const int* __restrict__ ei1, const float* __restrict__ ev1,
    const int* __restrict__ ei2, const float* __restrict__ ev2,
    const float* __restrict__ filt)
{
    const int c = blockIdx.y;
    const long task = (long)blockIdx.x * 16 + (threadIdx.x >> 4);
    const int lane16 = threadIdx.x & 15;
    if (task >= 3L * E_EDGES) return;
    const int seg = (int)(task / E_EDGES);
    const int e   = (int)(task % E_EDGES);
    const int*   ei = (seg == 0) ? ei0 : (seg == 1) ? ei1 : ei2;
    const float* ev = (seg == 0) ? ev0 : (seg == 1) ? ev1 : ev2;
    const int r  = ei[e];             // edge_index[0][e]
    const int cl = ei[E_EDGES + e];   // edge_index[1][e]
    const float w = ev[e] * filt[c * T_TYPES + seg];

    const float4 src = *(const float4*)(H + ((long)c * N_NODES + cl) * 64 + lane16 * 4);
    float* dst = Hout + ((long)c * N_NODES + r) * 64 + lane16 * 4;
    __hip_atomic_fetch_add(dst + 0, w * src.x, __ATOMIC_RELAXED, __HIP_MEMORY_SCOPE_AGENT);
    __hip_atomic_fetch_add(dst + 1, w * src.y, __ATOMIC_RELAXED, __HIP_MEMORY_SCOPE_AGENT);
    __hip_atomic_fetch_add(dst + 2, w * src.z, __ATOMIC_RELAXED, __HIP_MEMORY_SCOPE_AGENT);
    __hip_atomic_fetch_add(dst + 3, w * src.w, __ATOMIC_RELAXED, __HIP_MEMORY_SCOPE_AGENT);
}

// ---------------------------------------------------------------------------
// Hm[n][c*64+f] = TP*relu(BETA*X0 + (1-BETA)*H) + (1-TP)*X0
// ---------------------------------------------------------------------------
__global__ __launch_bounds__(256) void merge_kernel(
    const float* __restrict__ X0, const float* __restrict__ H,
    float* __restrict__ Hm)
{
    const long i = (long)blockIdx.x * blockDim.x + threadIdx.x;   // over C*N*64
    if (i >= (long)C_CH * N_NODES * 64) return;
    const int  f  = (int)(i & 63);
    const long nc = i >> 6;
    const int  n  = (int)(nc % N_NODES);
    const int  c  = (int)(nc / N_NODES);
    const float x0 = X0[i], h = H[i];
    const float v = TPROB * fmaxf(BETA * x0 + (1.0f - BETA) * h, 0.0f)
                  + (1.0f - TPROB) * x0;
    Hm[(long)n * (C_CH * 64) + c * 64 + f] = v;
}

// ---------------------------------------------------------------------------
extern "C" void kernel_launch(void* const* d_in, const int* in_sizes, int n_in,
                              void* d_out, int out_size, void* d_ws, size_t ws_size,
                              hipStream_t stream) {
    (void)in_sizes; (void)n_in; (void)out_size; (void)ws_size;
    const float* X     = (const float*)d_in[0];
    const float* ev0   = (const float*)d_in[1];
    const float* ev1   = (const float*)d_in[2];
    const float* ev2   = (const float*)d_in[3];
    const float* Ws0   = (const float*)d_in[4];
    const float* Ws1   = (const float*)d_in[5];
    const float* lw0   = (const float*)d_in[6];
    const float* lw1   = (const float*)d_in[7];
    const float* linW0 = (const float*)d_in[8];
    const float* linb0 = (const float*)d_in[9];
    const float* linW1 = (const float*)d_in[10];
    const float* linb1 = (const float*)d_in[11];
    const int*   ei0   = (const int*)d_in[12];
    const int*   ei1   = (const int*)d_in[13];
    const int*   ei2   = (const int*)d_in[14];
    float* out = (float*)d_out;

    const long CNH = (long)C_CH * N_NODES * 64;   // 6.4M floats
    float* ws   = (float*)d_ws;
    float* H    = ws;                             // [C][N][64]
    float* Hn   = H  + CNH;                       // [C][N][64]
    float* X0   = Hn + CNH;                       // [C][N][64]
    float* Hm   = X0 + CNH;                       // [N][128]
    float* B1   = Hm + CNH;                       // [N][64]  block-1 output
    float* filt = B1 + (long)N_NODES * 64;        // 24 floats

    filt_kernel<<<1, 32, 0, stream>>>(lw0, lw1, filt);

    const int tilesM     = N_NODES / 16;          // 3125 (exact)
    const int gemmBlocks = (tilesM + 7) / 8;      // 8 waves / 256-thread block

    dim3 spmmGrid((unsigned)((3L * E_EDGES + 15) / 16), C_CH);
    const int mergeBlocks = (int)((CNH + 255) / 256);

    for (int b = 0; b < 2; ++b) {
        const float* Xin = (b == 0) ? X : B1;
        const float* Ws  = (b == 0) ? Ws0 : Ws1;
        const float* lW  = (b == 0) ? linW0 : linW1;
        const float* lb  = (b == 0) ? linb0 : linb1;
        float* Ob        = (b == 0) ? B1 : out;
        const float* fb  = filt + b * NL * C_CH * T_TYPES;

        // Projection: H[c] = Xin @ Ws[c], X0 = detached copy of H
        gemm_n64_f32_wmma<<<dim3(gemmBlocks, C_CH), 256, 0, stream>>>(
            Xin, 64, Ws, (long)64 * 64, H, X0, (long)N_NODES * 64,
            nullptr, 0, N_NODES, 64);

        // NL sparse propagation passes (ping-pong H <-> Hn)
        float* Hc = H; float* Hd = Hn;
        for (int l = 0; l < NL; ++l) {
            hipMemsetAsync(Hd, 0, CNH * sizeof(float), stream);
            spmm_kernel<<<spmmGrid, 256, 0, stream>>>(
                Hc, Hd, ei0, ev0, ei1, ev1, ei2, ev2,
                fb + l * C_CH * T_TYPES);
            float* t = Hc; Hc = Hd; Hd = t;
        }

        // Merge + channel-major reshape -> Hm [N, 128]
        merge_kernel<<<mergeBlocks, 256, 0, stream>>>(X0, Hc, Hm);

        // Linear: Ob = relu(Hm @ linW + linb)
        gemm_n64_f32_wmma<<<dim3(gemmBlocks, 1), 256, 0, stream>>>(
            Hm, 128, lW, 0, Ob, nullptr, 0, lb, 1, N_NODES, 128);
    }
}